// MaxFeatBlockDescriptorLayer_71098888618039
// MI455X (gfx1250) — compile-verified
//
#include <hip/hip_runtime.h>
#include <hip/hip_bf16.h>

// ---------------------------------------------------------------------------
// MaxFeatBlockDescriptor for MI455X (gfx1250, wave32)
//
// emb:  [B=2, 128, 128, C=2048] f32      prob: [B=2, 128, 128, K=8] f32
// out:  [B=2, K=8, C=2048] f32
//
// Phase 1: per-class argmax + sum over N=16384 positions (1 MB scan).
// Phase 2: mask = (mean > 0.3); gather emb row, scale, store (256 KB).
//
// Argmax uses monotonic-key packing (order-preserving float->u32 transform,
// packed with ~index into u64) so every reduction stage is a branchless
// 64-bit max.  The cross-wave sum combine uses V_WMMA_F32_16X16X4_F32
// (ones-matrix reduction) to exercise the CDNA5 matrix path.
// ---------------------------------------------------------------------------

#define NPOS   16384
#define NCLS   8
#define NCHAN  2048
#define SLICES 16                 // slices per batch image
#define SLICE_LEN (NPOS / SLICES) // 1024
#define TAU    0.3f
#define NEG_BIG (-3.402823466e38f)

typedef __attribute__((ext_vector_type(2))) float v2f;
typedef __attribute__((ext_vector_type(8))) float v8f;
typedef unsigned long long u64;

// Order-preserving float -> u32 key: for any a<b (as floats), key(a)<key(b).
__device__ __forceinline__ unsigned mono_key(float v) {
    unsigned u = __float_as_uint(v);
    return u ^ ((unsigned)((int)u >> 31) | 0x80000000u);
}

// Pack (value, index) so that u64 max == (larger value, then smaller index).
__device__ __forceinline__ u64 pack_key(float v, int n) {
    return ((u64)mono_key(v) << 32) | (unsigned)(~n);
}

// ---------------------------------------------------------------------------
// Kernel 1: per-slice partial reduction (sum + argmax) for all 8 classes.
// grid = B * SLICES = 32 blocks, block = 256 threads (8 wave32s).
// ---------------------------------------------------------------------------
__global__ void mfbd_reduce_partial(const float* __restrict__ prob,
                                    float* __restrict__ bsum,   // [32][8]
                                    u64*   __restrict__ bkey)   // [32][8]
{
    const int blk   = blockIdx.x;       // 0..31
    const int b     = blk >> 4;
    const int slice = blk & (SLICES - 1);
    const int tid   = threadIdx.x;
    const int lane  = tid & 31;
    const int wave  = tid >> 5;

    const float* p = prob + (size_t)b * NPOS * NCLS;
    const int n0 = slice * SLICE_LEN;

    float sum[NCLS], mx[NCLS];
    int   mi[NCLS];
#pragma unroll
    for (int k = 0; k < NCLS; ++k) { sum[k] = 0.0f; mx[k] = NEG_BIG; mi[k] = 0; }

    // Strided scan: 8 contiguous floats (32 B) per position -> two b128 loads.
    // 4 iterations per thread; strictly-greater compare keeps the earliest n
    // per thread (n ascends), matching jnp.argmax first-occurrence semantics.
    for (int n = n0 + tid; n < n0 + SLICE_LEN; n += 256) {
        const float4* row = (const float4*)(p + (size_t)n * NCLS);
        float4 a = row[0];
        float4 c = row[1];
        float v[NCLS] = { a.x, a.y, a.z, a.w, c.x, c.y, c.z, c.w };
#pragma unroll
        for (int k = 0; k < NCLS; ++k) {
            sum[k] += v[k];
            if (v[k] > mx[k]) { mx[k] = v[k]; mi[k] = n; }
        }
    }

    // Pack once, then all further argmax reduction is branchless u64 max.
    u64 key[NCLS];
#pragma unroll
    for (int k = 0; k < NCLS; ++k) key[k] = pack_key(mx[k], mi[k]);

    // Intra-wave (wave32) butterfly reduction.
#pragma unroll
    for (int off = 16; off > 0; off >>= 1) {
#pragma unroll
        for (int k = 0; k < NCLS; ++k) {
            sum[k] += __shfl_xor(sum[k], off, 32);
            u64 ok = __shfl_xor(key[k], off, 32);
            key[k] = (ok > key[k]) ? ok : key[k];
        }
    }

    // Cross-wave combine via LDS (8 waves).
    __shared__ float s_sum[8][NCLS];
    __shared__ u64   s_key[8][NCLS];
    if (lane == 0) {
#pragma unroll
        for (int k = 0; k < NCLS; ++k) {
            s_sum[wave][k] = sum[k];
            s_key[wave][k] = key[k];
        }
    }
    __syncthreads();

    if (wave == 0) {  // wave-uniform branch: EXEC all-1s inside, required for WMMA
        // --- Sum combine across the 8 waves with two V_WMMA_F32_16X16X4_F32 ---
        // D = ones(16x4) x B(4x16) + C  => every row of D holds the K-dim column sums.
        // B layout (32-bit, 4x16): VGPR0 = K0 (lanes 0-15) / K2 (lanes 16-31),
        //                          VGPR1 = K1 (lanes 0-15) / K3 (lanes 16-31).
        // Column n = class (classes 8..15 padded with zero).
        const int  cls = lane & 15;
        const bool hi  = lane >= 16;
        float b0 = 0.0f, b1 = 0.0f, b2 = 0.0f, b3 = 0.0f;
        if (cls < NCLS) {
            b0 = s_sum[hi ? 2 : 0][cls];   // waves 0..3 -> first WMMA
            b1 = s_sum[hi ? 3 : 1][cls];
            b2 = s_sum[hi ? 6 : 4][cls];   // waves 4..7 -> second WMMA
            b3 = s_sum[hi ? 7 : 5][cls];
        }
        v2f A  = { 1.0f, 1.0f };
        v2f B0 = { b0, b1 };
        v2f B1 = { b2, b3 };
        v8f acc = {};
        acc = __builtin_amdgcn_wmma_f32_16x16x4_f32(false, A, false, B0,
                                                    (short)0, acc, false, false);
        acc = __builtin_amdgcn_wmma_f32_16x16x4_f32(false, A, false, B1,
                                                    (short)0, acc, false, false);
        // D VGPR0: lane L holds D[M=(L<16?0:8)][N=L%16]; rows are identical,
        // so acc[0] at lane L is the total slice sum for class L%16.
        float total = acc[0];

        if (lane < NCLS) {
            bsum[blk * NCLS + lane] = total;

            // --- Argmax combine across 8 waves: branchless u64 max ---
            u64 best = s_key[0][lane];
#pragma unroll
            for (int w = 1; w < 8; ++w) {
                u64 kk = s_key[w][lane];
                best = (kk > best) ? kk : best;
            }
            bkey[blk * NCLS + lane] = best;
        }
    }
}

// ---------------------------------------------------------------------------
// Kernel 2: final cross-slice reduction + masked gather of the emb row.
// grid = B * K = 16 blocks, block = 256 threads.
// ---------------------------------------------------------------------------
__global__ void mfbd_finalize_gather(const float* __restrict__ emb,
                                     const float* __restrict__ bsum,
                                     const u64*   __restrict__ bkey,
                                     float* __restrict__ out)
{
    const int m   = blockIdx.x;       // b*8 + k
    const int b   = m >> 3;
    const int k   = m & 7;
    const int tid = threadIdx.x;

    __shared__ int   s_n;
    __shared__ float s_scale;

    if (tid < 32) {
        float sv = 0.0f;
        u64   kv = 0;                 // smaller than any real packed key
        if (tid < SLICES) {
            const int off = (b * SLICES + tid) * NCLS + k;
            sv = bsum[off];
            kv = bkey[off];
        }
#pragma unroll
        for (int o = 8; o > 0; o >>= 1) {
            sv += __shfl_down(sv, o, 32);
            u64 ok = __shfl_down(kv, o, 32);
            kv = (ok > kv) ? ok : kv;
        }
        if (tid == 0) {
            s_n     = (int)(~(unsigned)(kv & 0xFFFFFFFFu));   // decode index
            s_scale = (sv * (1.0f / (float)NPOS) > TAU) ? 1.0f : 0.0f;
        }
    }
    __syncthreads();

    const int   n = s_n;
    const float s = s_scale;
    const float4* src = (const float4*)(emb + ((size_t)b * NPOS + n) * NCHAN);
    float4*       dst = (float4*)(out + (size_t)m * NCHAN);

    // 2048 floats = 512 float4; 2 per thread, fully coalesced b128.
#pragma unroll
    for (int i = tid; i < NCHAN / 4; i += 256) {
        float4 v = src[i];
        v.x *= s; v.y *= s; v.z *= s; v.w *= s;
        dst[i] = v;
    }
}

// ---------------------------------------------------------------------------
extern "C" void kernel_launch(void* const* d_in, const int* in_sizes, int n_in,
                              void* d_out, int out_size, void* d_ws, size_t ws_size,
                              hipStream_t stream)
{
    const float* emb  = (const float*)d_in[0];   // [2,128,128,2048] f32
    const float* prob = (const float*)d_in[1];   // [2,128,128,8] f32
    float* out = (float*)d_out;                  // [2,8,2048] f32

    // Workspace layout: bsum[32][8] f32 | bkey[32][8] u64 (8-byte aligned)
    float* bsum = (float*)d_ws;
    u64*   bkey = (u64*)((char*)d_ws + ((32 * NCLS * sizeof(float) + 15) & ~15));

    mfbd_reduce_partial<<<2 * SLICES, 256, 0, stream>>>(prob, bsum, bkey);
    mfbd_finalize_gather<<<2 * NCLS, 256, 0, stream>>>(emb, bsum, bkey, out);
}